// CenteredBilinearPooling2_34213709480492
// MI455X (gfx1250) — compile-verified
//
#include <hip/hip_runtime.h>

typedef float v2f __attribute__((ext_vector_type(2)));
typedef float v4f __attribute__((ext_vector_type(4)));
typedef float v8f __attribute__((ext_vector_type(8)));

#define T_DIM   2048
#define C_DIM   128
#define L_DIM   15
#define PADW    7
#define WIN     29      // x rows needed: [s-14, s+14]
#define TP_DIM  1024
#define VSTR    132     // LDS row stride for VA/VB (bank-conflict avoidance)
#define OUT_ROW (C_DIM + C_DIM * C_DIM)   // 16512

__global__ __launch_bounds__(256)
void cbp_wmma_kernel(const float* __restrict__ x,
                     const float* __restrict__ w1,
                     const float* __restrict__ w2,
                     float* __restrict__ out)
{
    __shared__ float xs[WIN][C_DIM];     // x window, zero-padded
    __shared__ float VA[16][VSTR];       // w2[l] * xc  (K rows, padded to 16)
    __shared__ float VB[16][VSTR];       // xc
    __shared__ float lw1[L_DIM];
    __shared__ float lw2[L_DIM];

    const int tp  = blockIdx.x;          // 0..1023
    const int b   = blockIdx.y;          // 0..3
    const int s   = tp * 2;
    const int tid = threadIdx.x;

    if (tid < L_DIM) { lw1[tid] = w1[tid]; lw2[tid] = w2[tid]; }

    // ---- Stage 1: load x[b, s-14 .. s+14, :] into LDS (float4, zero-pad OOB)
    const float* xb = x + (size_t)b * T_DIM * C_DIM;
    for (int idx = tid; idx < WIN * (C_DIM / 4); idx += 256) {
        int row = idx >> 5;              // C_DIM/4 == 32 quads per row
        int c4  = (idx & 31) << 2;
        int t   = s - 14 + row;
        v4f v = {0.f, 0.f, 0.f, 0.f};
        if (t >= 0 && t < T_DIM)
            v = *(const v4f*)(xb + (size_t)t * C_DIM + c4);
        *(v4f*)(&xs[row][c4]) = v;
    }
    __syncthreads();

    // ---- Stage 2: moving average mu, centered window V, and mu output
    const size_t obase = ((size_t)b * TP_DIM + tp) * (size_t)OUT_ROW;
    for (int idx = tid; idx < 16 * C_DIM; idx += 256) {
        int l = idx >> 7;                // 0..15 (row 15 = K padding, zeros)
        int c = idx & 127;
        float va = 0.f, vb = 0.f;
        if (l < L_DIM) {
            float mu = 0.f;
            #pragma unroll
            for (int j = 0; j < L_DIM; ++j)
                mu += lw1[j] * xs[l + j][c];
            int t = s - PADW + l;
            float v = (t >= 0 && t < T_DIM) ? (xs[l + PADW][c] - mu) : 0.f;
            va = lw2[l] * v;
            vb = v;
            if (l == PADW)               // t == s: mu part of the output row
                out[obase + c] = mu;
        }
        VA[l][c] = va;
        VB[l][c] = vb;
    }
    __syncthreads();

    // ---- Stage 3: sigma = VA^T * VB via V_WMMA_F32_16X16X4_F32, K = 16
    // 8 waves: wave w owns M-tile w (rows c0..c0+15), loops over 8 N-tiles.
    const int wave = tid >> 5;
    const int lane = tid & 31;
    const int half = lane >> 4;          // 0: K={k0,k0+1}  1: K={k0+2,k0+3}
    const int mn   = lane & 15;
    const int c0   = wave << 4;

    // A fragments (reused for all 8 N-tiles): A'[c, k] = VA[k][c]
    v2f afrag[4];
    #pragma unroll
    for (int kk = 0; kk < 4; ++kk) {
        int k = 4 * kk + 2 * half;
        afrag[kk].x = VA[k    ][c0 + mn];
        afrag[kk].y = VA[k + 1][c0 + mn];
    }

    float* osig = out + obase + C_DIM;
    for (int dt = 0; dt < 8; ++dt) {
        int n0 = dt << 4;
        v8f acc = {0.f, 0.f, 0.f, 0.f, 0.f, 0.f, 0.f, 0.f};
        #pragma unroll
        for (int kk = 0; kk < 4; ++kk) {
            int k = 4 * kk + 2 * half;
            v2f bfrag;
            bfrag.x = VB[k    ][n0 + mn];
            bfrag.y = VB[k + 1][n0 + mn];
            acc = __builtin_amdgcn_wmma_f32_16x16x4_f32(
                      false, afrag[kk], false, bfrag,
                      (short)0, acc, false, false);
        }
        // D layout: VGPR r -> row m = r + 8*half, col n0 + mn
        #pragma unroll
        for (int r = 0; r < 8; ++r) {
            int m = r + 8 * half;
            osig[(size_t)(c0 + m) * C_DIM + n0 + mn] = acc[r];
        }
    }
}

extern "C" void kernel_launch(void* const* d_in, const int* in_sizes, int n_in,
                              void* d_out, int out_size, void* d_ws, size_t ws_size,
                              hipStream_t stream) {
    const float* x  = (const float*)d_in[0];
    const float* w1 = (const float*)d_in[1];
    const float* w2 = (const float*)d_in[2];
    float* out = (float*)d_out;

    dim3 grid(TP_DIM, 4, 1);   // (Tp, B)
    dim3 block(256, 1, 1);
    cbp_wmma_kernel<<<grid, block, 0, stream>>>(x, w1, w2, out);
}